// CombinedNMS_63668595196393
// MI455X (gfx1250) — compile-verified
//
#include <hip/hip_runtime.h>
#include <hip/hip_bf16.h>
#include <math.h>

#define NBOX   262144
#define NCLS   90
#define NEGV   (-1e30f)
#define IOU_T  0.35f
#define RBLK   1024      // blocks for suppress+reduce pass (256 thr each -> covers NBOX)
#define STB    128       // boxes per scoring block (128*90*4B = 45 KB LDS tile)

// ---------------- K1: xywh -> rel yxyx, areas; init curbox.valid and ticket ----------------
__global__ void nms_prep(const float* __restrict__ box, const float* __restrict__ ishape,
                         float* __restrict__ byxyx, float* __restrict__ areas,
                         float* __restrict__ curbox, unsigned* __restrict__ ticket)
{
    int j = blockIdx.x * blockDim.x + threadIdx.x;
    if (blockIdx.x == 0 && threadIdx.x == 0) { curbox[5] = 0.0f; *ticket = 0u; }
    float h = ishape[0], w = ishape[1];
    float4 b = reinterpret_cast<const float4*>(box)[j];          // x, y, bw, bh
    float y1 = b.y / h;
    float x1 = b.x / w;
    float y2 = (b.y + b.w) / h;                                  // (y + bh)/h
    float x2 = (b.x + b.z) / w;                                  // (x + bw)/w
    reinterpret_cast<float4*>(byxyx)[j] = make_float4(y1, x1, y2, x2);
    areas[j] = (y2 - y1) * (x2 - x1);
}

// ---------------- K2: scores/labels via async global->LDS staged tile ----------------
// Each block stages a contiguous 128x90 float tile (45 KB) into LDS with
// global_load_async_to_lds_b128 (coalesced 512B per wave-instruction), waits on
// ASYNCcnt, then each thread reduces its 90-class row from LDS.
// sigmoid is monotone: max/argmax taken over raw logits, one expf per box.
__global__ void nms_score(const float* __restrict__ cls, float* __restrict__ scores,
                          int* __restrict__ labels)
{
    __shared__ float tile[STB * NCLS];                           // 46080 B
    const int tid = threadIdx.x;
    const long long base = (long long)blockIdx.x * STB;
    const float* g = cls + base * NCLS;

    for (int i = tid; i < (STB * NCLS) / 4; i += STB) {
        unsigned lds = (unsigned)(size_t)(void*)&tile[i * 4];    // LDS byte offset
        const float* gp = g + (size_t)i * 4;
        asm volatile("global_load_async_to_lds_b128 %0, %1, off"
                     :: "v"(lds), "v"(gp) : "memory");
    }
#if __has_builtin(__builtin_amdgcn_s_wait_asynccnt)
    __builtin_amdgcn_s_wait_asynccnt(0);
#else
    asm volatile("s_wait_asynccnt 0" ::: "memory");
#endif
    __syncthreads();

    float m = tile[tid * NCLS];
    int a = 0;
    for (int c = 1; c < NCLS; ++c) {                             // first-max (ties -> low c)
        float v = tile[tid * NCLS + c];
        if (v > m) { m = v; a = c; }
    }
    scores[base + tid] = 1.0f / (1.0f + expf(-m));               // sigmoid(max logit)
    labels[base + tid] = a;
}

// ---------------- K3: one fused NMS iteration ----------------
// Phase 1 (all blocks): suppress vs. previously selected box, block argmax -> partials.
// Phase 2 (last-ticket block): reduce all partials, record selection, update curbox,
// scores[sel] = NEG, reset ticket for the next dispatch.
__global__ void nms_iter(const float* __restrict__ byxyx, const float* __restrict__ areas,
                         float* __restrict__ scores, float* __restrict__ curbox,
                         float* __restrict__ pmax, int* __restrict__ pidx,
                         unsigned* __restrict__ ticket,
                         int* __restrict__ selIdx, float* __restrict__ selScore,
                         int* __restrict__ selValid, int iter)
{
    const int t = threadIdx.x;
    const int j = blockIdx.x * blockDim.x + t;

    float cy1 = curbox[0], cx1 = curbox[1], cy2 = curbox[2], cx2 = curbox[3];
    float carea = curbox[4], cvalid = curbox[5];

    float s = scores[j];
    if (cvalid > 0.5f) {
        float4 b = reinterpret_cast<const float4*>(byxyx)[j];    // y1,x1,y2,x2
        float yy1 = fmaxf(cy1, b.x);
        float xx1 = fmaxf(cx1, b.y);
        float yy2 = fminf(cy2, b.z);
        float xx2 = fminf(cx2, b.w);
        float inter = fmaxf(yy2 - yy1, 0.0f) * fmaxf(xx2 - xx1, 0.0f);
        float iou = inter / (carea + areas[j] - inter + 1e-9f);
        if (iou > IOU_T) { s = NEGV; scores[j] = NEGV; }
    }

    __shared__ float sv[256];
    __shared__ int   si[256];
    __shared__ unsigned isLast;
    sv[t] = s; si[t] = j;
    __syncthreads();
    for (int off = 128; off > 0; off >>= 1) {
        if (t < off) {
            float v = sv[t + off]; int i2 = si[t + off];
            if (v > sv[t] || (v == sv[t] && i2 < si[t])) { sv[t] = v; si[t] = i2; }
        }
        __syncthreads();
    }
    if (t == 0) {
        pmax[blockIdx.x] = sv[0];
        pidx[blockIdx.x] = si[0];
        __threadfence();                                         // release partials
        unsigned prev = atomicAdd(ticket, 1u);
        isLast = (prev == (unsigned)(gridDim.x - 1)) ? 1u : 0u;
    }
    __syncthreads();

    if (isLast) {
        __threadfence();                                         // acquire partials
        float best = pmax[t]; int bi = pidx[t];
        for (int k = 1; k < RBLK / 256; ++k) {
            float v = pmax[t + 256 * k]; int i2 = pidx[t + 256 * k];
            if (v > best || (v == best && i2 < bi)) { best = v; bi = i2; }
        }
        __syncthreads();                                         // sv/si reuse
        sv[t] = best; si[t] = bi;
        __syncthreads();
        for (int off = 128; off > 0; off >>= 1) {
            if (t < off) {
                float v = sv[t + off]; int i2 = si[t + off];
                if (v > sv[t] || (v == sv[t] && i2 < si[t])) { sv[t] = v; si[t] = i2; }
            }
            __syncthreads();
        }
        if (t == 0) {
            int idx = si[0]; float sbest = sv[0];
            int valid = (sbest > NEGV * 0.5f) ? 1 : 0;
            selIdx[iter] = idx; selScore[iter] = sbest; selValid[iter] = valid;
            float4 b = reinterpret_cast<const float4*>(byxyx)[idx];
            curbox[0] = b.x; curbox[1] = b.y; curbox[2] = b.z; curbox[3] = b.w;
            curbox[4] = areas[idx];
            curbox[5] = valid ? 1.0f : 0.0f;
            scores[idx] = NEGV;                                  // s.at[i].set(NEG), unconditional
            *ticket = 0u;                                        // re-arm for next dispatch
        }
    }
}

// ---------------- K4: format outputs (top_k over 32 is identity; scores are
// non-increasing across greedy iterations and invalid entries form a NEG suffix) ----------------
__global__ void nms_out(const int* __restrict__ selIdx, const float* __restrict__ selScore,
                        const int* __restrict__ selValid, const int* __restrict__ labels,
                        const float* __restrict__ byxyx, float* __restrict__ out)
{
    int t = threadIdx.x;
    if (t < 32) {
        int v = selValid[t];
        int idx = selIdx[t];
        float4 b = reinterpret_cast<const float4*>(byxyx)[idx];
        out[t * 4 + 0] = v ? b.x : 0.0f;
        out[t * 4 + 1] = v ? b.y : 0.0f;
        out[t * 4 + 2] = v ? b.z : 0.0f;
        out[t * 4 + 3] = v ? b.w : 0.0f;
        out[128 + t] = v ? (float)labels[idx] : -1.0f;
        out[160 + t] = v ? selScore[t] : 0.0f;
    }
    if (t == 0) {
        int c = 0;
        for (int i = 0; i < 32; ++i) c += selValid[i];
        out[192] = (float)c;
    }
}

extern "C" void kernel_launch(void* const* d_in, const int* in_sizes, int n_in,
                              void* d_out, int out_size, void* d_ws, size_t ws_size,
                              hipStream_t stream)
{
    (void)in_sizes; (void)n_in; (void)out_size; (void)ws_size;
    const float* box    = (const float*)d_in[0];   // (1, N, 4)
    const float* cls    = (const float*)d_in[1];   // (1, N, 90)
    const float* ishape = (const float*)d_in[2];   // (2,)
    float* out = (float*)d_out;                    // 193 floats

    // Workspace layout (~7.01 MB)
    char* w = (char*)d_ws;
    float*    byxyx    = (float*)w;     w += (size_t)NBOX * 4 * sizeof(float);
    float*    areas    = (float*)w;     w += (size_t)NBOX * sizeof(float);
    float*    scores   = (float*)w;     w += (size_t)NBOX * sizeof(float);
    int*      labels   = (int*)w;       w += (size_t)NBOX * sizeof(int);
    float*    pmax     = (float*)w;     w += RBLK * sizeof(float);
    int*      pidx     = (int*)w;       w += RBLK * sizeof(int);
    float*    curbox   = (float*)w;     w += 8 * sizeof(float);
    int*      selIdx   = (int*)w;       w += 32 * sizeof(int);
    float*    selScore = (float*)w;     w += 32 * sizeof(float);
    int*      selValid = (int*)w;       w += 32 * sizeof(int);
    unsigned* ticket   = (unsigned*)w;  w += sizeof(unsigned);

    nms_prep<<<NBOX / 256, 256, 0, stream>>>(box, ishape, byxyx, areas, curbox, ticket);
    nms_score<<<NBOX / STB, STB, 0, stream>>>(cls, scores, labels);
    for (int it = 0; it < 32; ++it) {
        nms_iter<<<RBLK, 256, 0, stream>>>(byxyx, areas, scores, curbox, pmax, pidx,
                                           ticket, selIdx, selScore, selValid, it);
    }
    nms_out<<<1, 64, 0, stream>>>(selIdx, selScore, selValid, labels, byxyx, out);
}